// OmicsIntegrationArch_17471926960174
// MI455X (gfx1250) — compile-verified
//
#include <hip/hip_runtime.h>
#include <hip/hip_bf16.h>

typedef float v2f __attribute__((ext_vector_type(2)));
typedef float v8f __attribute__((ext_vector_type(8)));

// ---------------------------------------------------------------- utilities

__global__ void zero_fill(float* __restrict__ p, long long total) {
    long long gid = (long long)blockIdx.x * blockDim.x + threadIdx.x;
    if (gid < total) p[gid] = 0.0f;
}

// ------------------------------------------------------- edge scatter (agg)
// agg[dst[e]][c] += x[src[e]][c]; one thread per (edge, channel).
// Adjacent lanes cover contiguous channels of the same edge -> coalesced
// loads and coalesced fp32 atomics that stay resident in L2.
__global__ void gin_scatter(const float* __restrict__ X,
                            const int* __restrict__ src,
                            const int* __restrict__ dst,
                            float* __restrict__ AGG,
                            int log2C, long long total) {
    long long gid = (long long)blockIdx.x * blockDim.x + threadIdx.x;
    if (gid >= total) return;
    int e = (int)(gid >> log2C);
    int c = (int)(gid & ((1 << log2C) - 1));
    int s = src[e];
    int d = dst[e];
    float v = X[((size_t)s << log2C) + c];
    __hip_atomic_fetch_add(&AGG[((size_t)d << log2C) + c], v,
                           __ATOMIC_RELAXED, __HIP_MEMORY_SCOPE_AGENT);
}

// ------------------------------------------------- WMMA GEMM: Y=(X+AGG)W^T+b
// One wave per 16x16 output tile, f32 WMMA 16x16x4, K stepped by 4.
// A-frag (2 VGPR): lane<16 -> K={k,k+1}, lane>=16 -> K={k+2,k+3}, M=lane&15.
// B-frag mirrors with N=lane&15 (B = W^T, so read W[row=outCh][k]).
// C/D (8 VGPR): c[r] = Y[row0 + r + 8*(lane>>4)][col0 + (lane&15)].
__global__ __launch_bounds__(256)
void gin_gemm_wmma(const float* __restrict__ X,
                   const float* __restrict__ AGG,
                   const float* __restrict__ W,     // [Cout, Cin] row-major
                   const float* __restrict__ bias,  // [Cout]
                   float* __restrict__ Y,           // [N, Cout]
                   int Cin, int Cout, int totalTiles, int ctiles, int relu) {
    int wave = threadIdx.x >> 5;
    int lane = threadIdx.x & 31;
    int tile = blockIdx.x * 8 + wave;
    if (tile >= totalTiles) return;              // wave-uniform: EXEC stays ~0

    int rowTile = tile / ctiles;
    int colTile = tile - rowTile * ctiles;
    int row0 = rowTile << 4;
    int col0 = colTile << 4;
    int laneM = lane & 15;
    int kh    = lane >> 4;                        // 0 or 1

    const float* __restrict__ arow = X   + (size_t)(row0 + laneM) * Cin;
    const float* __restrict__ grow = AGG + (size_t)(row0 + laneM) * Cin;
    const float* __restrict__ wrow = W   + (size_t)(col0 + laneM) * Cin;

    v8f c = {};
#pragma unroll 4
    for (int k = 0; k < Cin; k += 4) {
        int kk = k + 2 * kh;
        v2f a, b;
        a.x = arow[kk]     + grow[kk];
        a.y = arow[kk + 1] + grow[kk + 1];
        b.x = wrow[kk];
        b.y = wrow[kk + 1];
        c = __builtin_amdgcn_wmma_f32_16x16x4_f32(
                /*neg_a=*/false, a, /*neg_b=*/false, b,
                /*c_mod=*/(short)0, c, /*reuse_a=*/false, /*reuse_b=*/false);
    }

    float bv = bias[col0 + laneM];
#pragma unroll
    for (int r = 0; r < 8; ++r) {
        float v = c[r] + bv;
        if (relu) v = fmaxf(v, 0.0f);
        Y[(size_t)(row0 + r + 8 * kh) * Cout + col0 + laneM] = v;
    }
}

// ------------------------------------------------------------ BN statistics
// stats[0..C) = sum, stats[C..2C) = sum of squares (pre-zeroed).
__global__ __launch_bounds__(256)
void bn_stats(const float* __restrict__ H, float* __restrict__ stats,
              int n, int C) {
    int tid  = threadIdx.x;
    int c    = tid & (C - 1);          // C is 64 or 128 (power of two)
    int repl = tid / C;
    int replCount = 256 / C;
    float s = 0.0f, s2 = 0.0f;
    for (int r = blockIdx.x * replCount + repl; r < n;
         r += gridDim.x * replCount) {
        float v = H[(size_t)r * C + c];
        s += v;
        s2 += v * v;
    }
    __shared__ float ls[256];
    __shared__ float ls2[256];
    ls[tid] = s; ls2[tid] = s2;
    __syncthreads();
    if (repl == 0) {
        for (int i = 1; i < replCount; ++i) {
            s  += ls[i * C + c];
            s2 += ls2[i * C + c];
        }
        __hip_atomic_fetch_add(&stats[c], s,
                               __ATOMIC_RELAXED, __HIP_MEMORY_SCOPE_AGENT);
        __hip_atomic_fetch_add(&stats[C + c], s2,
                               __ATOMIC_RELAXED, __HIP_MEMORY_SCOPE_AGENT);
    }
}

// -------------------------------------------------- BN normalize + ReLU out
__global__ void bn_apply(const float* __restrict__ H,
                         const float* __restrict__ stats,
                         const float* __restrict__ g,
                         const float* __restrict__ beta,
                         float* __restrict__ Y, int n, int C) {
    long long gid = (long long)blockIdx.x * blockDim.x + threadIdx.x;
    if (gid >= (long long)n * C) return;
    int c = (int)(gid & (C - 1));
    float invN = 1.0f / (float)n;
    float m   = stats[c] * invN;
    float var = stats[C + c] * invN - m * m;
    float inv = rsqrtf(var + 1e-5f);
    float v = (H[gid] - m) * inv * g[c] + beta[c];
    Y[gid] = fmaxf(v, 0.0f);
}

// ------------------------------------------------------------- host helpers

static void run_gin_layer(const float* X, const int* edges, int E, int n,
                          int Cin, int Cout,
                          const float* W, const float* b,
                          const float* g, const float* beta,  // null -> ReLU only
                          float* AGG, float* H, float* stats,
                          float* OUT, hipStream_t stream) {
    // zero agg
    long long aggTot = (long long)n * Cin;
    zero_fill<<<(int)((aggTot + 255) / 256), 256, 0, stream>>>(AGG, aggTot);
    // scatter-add x[src] into agg[dst]
    long long scTot = (long long)E * Cin;
    int log2C = (Cin == 256) ? 8 : (Cin == 128) ? 7 : 6;
    gin_scatter<<<(int)((scTot + 255) / 256), 256, 0, stream>>>(
        X, edges, edges + E, AGG, log2C, scTot);
    // GEMM
    int ctiles = Cout >> 4;
    int totalTiles = (n >> 4) * ctiles;
    int blocks = (totalTiles + 7) / 8;
    if (g) {
        gin_gemm_wmma<<<blocks, 256, 0, stream>>>(X, AGG, W, b, H,
                                                  Cin, Cout, totalTiles, ctiles, 0);
        zero_fill<<<1, 2 * Cout, 0, stream>>>(stats, 2 * Cout);
        bn_stats<<<256, 256, 0, stream>>>(H, stats, n, Cout);
        long long tot = (long long)n * Cout;
        bn_apply<<<(int)((tot + 255) / 256), 256, 0, stream>>>(
            H, stats, g, beta, OUT, n, Cout);
    } else {
        gin_gemm_wmma<<<blocks, 256, 0, stream>>>(X, AGG, W, b, OUT,
                                                  Cin, Cout, totalTiles, ctiles, 1);
    }
}

extern "C" void kernel_launch(void* const* d_in, const int* in_sizes, int n_in,
                              void* d_out, int out_size, void* d_ws, size_t ws_size,
                              hipStream_t stream) {
    const float* ft0 = (const float*)d_in[0];
    const float* ft1 = (const float*)d_in[1];
    const float* fs  = (const float*)d_in[2];
    const int*   et0 = (const int*)d_in[3];
    const int*   et1 = (const int*)d_in[4];
    const int*   es  = (const int*)d_in[5];
    const float* aT0_W = (const float*)d_in[6],  *aT0_b = (const float*)d_in[7];
    const float* aT0_g = (const float*)d_in[8],  *aT0_be = (const float*)d_in[9];
    const float* aT1_W = (const float*)d_in[10], *aT1_b = (const float*)d_in[11];
    const float* aT1_g = (const float*)d_in[12], *aT1_be = (const float*)d_in[13];
    const float* aS_W  = (const float*)d_in[14], *aS_b  = (const float*)d_in[15];
    const float* aS_g  = (const float*)d_in[16], *aS_be = (const float*)d_in[17];
    const float* ext_W = (const float*)d_in[18], *ext_b = (const float*)d_in[19];
    const float* ext_g = (const float*)d_in[20], *ext_be = (const float*)d_in[21];
    const float* rT0_W = (const float*)d_in[22], *rT0_b = (const float*)d_in[23];
    const float* rT1_W = (const float*)d_in[24], *rT1_b = (const float*)d_in[25];

    const int N = in_sizes[0] / 256;
    const int E = in_sizes[3] / 2;

    // workspace layout (floats)
    float* ws    = (float*)d_ws;
    float* AGG   = ws;                                  // N*256
    float* A0    = AGG + (size_t)N * 256;               // N*128
    float* A1    = A0  + (size_t)N * 128;               // N*128
    float* AS    = A1  + (size_t)N * 128;               // N*128
    float* HTMP  = AS  + (size_t)N * 128;               // N*128 (pre-BN)
    float* STATS = HTMP + (size_t)N * 128;              // 512

    // output layout (floats): hs[N*64] | ht[2N*64] | f0[N*256] | f1[N*128]
    float* out = (float*)d_out;
    float* HS = out;
    float* H0 = out + (size_t)N * 64;
    float* H1 = H0  + (size_t)N * 64;
    float* F0 = H1  + (size_t)N * 64;
    float* F1 = F0  + (size_t)N * 256;

    // align layers (Linear + BN + ReLU)
    run_gin_layer(ft0, et0, E, N, 256, 128, aT0_W, aT0_b, aT0_g, aT0_be,
                  AGG, HTMP, STATS, A0, stream);
    run_gin_layer(ft1, et1, E, N, 128, 128, aT1_W, aT1_b, aT1_g, aT1_be,
                  AGG, HTMP, STATS, A1, stream);
    run_gin_layer(fs,  es,  E, N, 256, 128, aS_W,  aS_b,  aS_g,  aS_be,
                  AGG, HTMP, STATS, AS, stream);
    // shared extractor (Linear + BN + ReLU)
    run_gin_layer(A0, et0, E, N, 128, 64, ext_W, ext_b, ext_g, ext_be,
                  AGG, HTMP, STATS, H0, stream);
    run_gin_layer(A1, et1, E, N, 128, 64, ext_W, ext_b, ext_g, ext_be,
                  AGG, HTMP, STATS, H1, stream);
    run_gin_layer(AS, es,  E, N, 128, 64, ext_W, ext_b, ext_g, ext_be,
                  AGG, HTMP, STATS, HS, stream);
    // recon layers (Linear + ReLU, no BN)
    run_gin_layer(H0, et0, E, N, 64, 256, rT0_W, rT0_b, nullptr, nullptr,
                  AGG, HTMP, STATS, F0, stream);
    run_gin_layer(H1, et1, E, N, 64, 128, rT1_W, rT1_b, nullptr, nullptr,
                  AGG, HTMP, STATS, F1, stream);
}